// ExternalAttention_56470230007869
// MI455X (gfx1250) — compile-verified
//
#include <hip/hip_runtime.h>

typedef float v2f __attribute__((ext_vector_type(2)));
typedef float v8f __attribute__((ext_vector_type(8)));

#define D_MODEL 512
#define N_HEADS 8
#define HEAD_D  64
#define BATCH   8
#define SEQ     8192
#define POI_NN  1683
#define POI_FF  24

// workspace layout (float offsets)
#define WS_Q      0         // 512
#define WS_QBIAS  512       // 16
#define WS_WQK    1024      // 512*16 = 8192 (padded columns 8..15 = 0)
#define WS_SCORES 16384     // 8*8*8192 = 524288
#define WS_XA     540672    // 8*8*512  = 32768
#define WS_OUTROW 573440    // 8*512    = 4096
// total = 577536 floats ~= 2.31 MB

// ---------------------------------------------------------------------------
// K1: q = ((poi @ wq1 + bq1) @ wq2 + bq2); wqk[c][h] = sum_e wk[c,64h+e]*q[h,e]
//     qbias[h] = sum_e bk[64h+e]*q[h,e]
// ---------------------------------------------------------------------------
__global__ __launch_bounds__(256)
void k_prepare(const float* __restrict__ poi, const float* __restrict__ wq1,
               const float* __restrict__ bq1, const float* __restrict__ wq2,
               const float* __restrict__ bq2, const float* __restrict__ wk,
               const float* __restrict__ bk, float* __restrict__ ws) {
    __shared__ float qv[POI_NN];
    __shared__ float q[D_MODEL];
    const int tid = threadIdx.x;

    for (int i = tid; i < POI_NN; i += 256) {
        float acc = bq1[0];
#pragma unroll
        for (int f = 0; f < POI_FF; ++f) acc += poi[i * POI_FF + f] * wq1[f];
        qv[i] = acc;
    }
    __syncthreads();

    for (int dcol = tid; dcol < D_MODEL; dcol += 256) {
        float acc = bq2[dcol];
        for (int i = 0; i < POI_NN; ++i) acc += qv[i] * wq2[i * D_MODEL + dcol];
        q[dcol]       = acc;
        ws[WS_Q + dcol] = acc;
    }
    __syncthreads();

    if (tid < 16) {
        float acc = 0.0f;
        if (tid < N_HEADS) {
#pragma unroll 8
            for (int e = 0; e < HEAD_D; ++e)
                acc += bk[tid * HEAD_D + e] * q[tid * HEAD_D + e];
        }
        ws[WS_QBIAS + tid] = acc;
    }

    for (int c = tid; c < D_MODEL; c += 256) {
#pragma unroll
        for (int h = 0; h < N_HEADS; ++h) {
            float acc = 0.0f;
            for (int e = 0; e < HEAD_D; ++e)
                acc += wk[c * D_MODEL + h * HEAD_D + e] * q[h * HEAD_D + e];
            ws[WS_WQK + c * 16 + h] = acc;
        }
#pragma unroll
        for (int h = N_HEADS; h < 16; ++h) ws[WS_WQK + c * 16 + h] = 0.0f;
    }
}

// ---------------------------------------------------------------------------
// K2: scores[b,h,s] = x[b,s,:] . wqk[:,h] + qbias[h]
//     WMMA f32 16x16x4: each wave computes a 16-row x 16-col (8 valid) tile,
//     K = 512 in steps of 4 -> 128 v_wmma_f32_16x16x4_f32 per wave.
// ---------------------------------------------------------------------------
__global__ __launch_bounds__(256)
void k_scores(const float* __restrict__ x, float* __restrict__ ws) {
    __shared__ float wqk[D_MODEL * 16];   // 32 KB, B-matrix [k][n] padded
    const int tid = threadIdx.x;
    const float* wsrc = ws + WS_WQK;
    for (int i = tid; i < D_MODEL * 16; i += 256) wqk[i] = wsrc[i];
    __syncthreads();

    const int wave    = tid >> 5;
    const int lane    = tid & 31;
    const int tile    = blockIdx.x * 8 + wave;   // 4096 tiles of 16 rows
    const int rowBase = tile * 16;               // global row in [0, 65536)
    const int m       = lane & 15;               // A row / B,N column for this lane
    const int khalf   = lane >> 4;               // which K-pair (0->K0,1 ; 1->K2,3)

    const float* xrow = x + (size_t)(rowBase + m) * D_MODEL;

    v8f c = {0.f, 0.f, 0.f, 0.f, 0.f, 0.f, 0.f, 0.f};
    for (int k0 = 0; k0 < D_MODEL; k0 += 4) {
        const int kk = k0 + khalf * 2;
        v2f a, bf;
        a.x  = xrow[kk];
        a.y  = xrow[kk + 1];
        bf.x = wqk[kk * 16 + m];
        bf.y = wqk[(kk + 1) * 16 + m];
        c = __builtin_amdgcn_wmma_f32_16x16x4_f32(
                /*neg_a=*/false, a, /*neg_b=*/false, bf,
                /*c_mod=*/(short)0, c, /*reuse_a=*/false, /*reuse_b=*/false);
    }

    // C/D layout: VGPR r -> row (r + 8*khalf), col = lane&15
    const int n = m;
    if (n < N_HEADS) {
        const float bias = ws[WS_QBIAS + n];
        float* sc = ws + WS_SCORES;
#pragma unroll
        for (int r = 0; r < 8; ++r) {
            const int row  = rowBase + r + khalf * 8;
            const int bidx = row >> 13;           // row / SEQ
            const int s    = row & (SEQ - 1);
            sc[((size_t)bidx * N_HEADS + n) * SEQ + s] = c[r] + bias;
        }
    }
}

// ---------------------------------------------------------------------------
// K3: in-place softmax over s for each (b,h); also zeroes the xa accumulator.
// ---------------------------------------------------------------------------
__global__ __launch_bounds__(256)
void k_softmax(float* __restrict__ ws) {
    __shared__ float red[256];
    const int tid = threadIdx.x;
    const int bh  = blockIdx.x;                  // 64 = B*H

    // zero xa (32768 floats / 16384 threads = 2 each)
    {
        const int g = bh * 256 + tid;
        ws[WS_XA + g]         = 0.0f;
        ws[WS_XA + 16384 + g] = 0.0f;
    }

    float* row = ws + WS_SCORES + (size_t)bh * SEQ;
    const float scale = 0.125f;                  // 1/sqrt(64)

    float mx = -3.402823466e38f;
    for (int s = tid; s < SEQ; s += 256) mx = fmaxf(mx, scale * row[s]);
    red[tid] = mx; __syncthreads();
    for (int off = 128; off > 0; off >>= 1) {
        if (tid < off) red[tid] = fmaxf(red[tid], red[tid + off]);
        __syncthreads();
    }
    mx = red[0];
    __syncthreads();

    float sum = 0.0f;
    for (int s = tid; s < SEQ; s += 256) {
        const float e = __expf(scale * row[s] - mx);
        row[s] = e;
        sum += e;
    }
    red[tid] = sum; __syncthreads();
    for (int off = 128; off > 0; off >>= 1) {
        if (tid < off) red[tid] += red[tid + off];
        __syncthreads();
    }
    const float inv = 1.0f / red[0];
    for (int s = tid; s < SEQ; s += 256) row[s] *= inv;
}

// ---------------------------------------------------------------------------
// K4: xa[b,h,c] += sum_s A[b,h,s] * x[b,s,c]   (s split 8 ways, f32 atomics)
// ---------------------------------------------------------------------------
__global__ __launch_bounds__(256)
void k_xa(const float* __restrict__ x, float* __restrict__ ws) {
    __shared__ float probs[N_HEADS * 1024];      // 32 KB
    const int b   = blockIdx.x;                  // 8
    const int ct  = blockIdx.y;                  // 8 column tiles of 64
    const int ssp = blockIdx.z;                  // 8 s-splits of 1024
    const int tid = threadIdx.x;
    const int s0  = ssp * 1024;

    const float* A = ws + WS_SCORES;
    for (int i = tid; i < N_HEADS * 1024; i += 256) {
        const int h = i >> 10, off = i & 1023;
        probs[i] = A[((size_t)b * N_HEADS + h) * SEQ + s0 + off];
    }
    __syncthreads();

    const int c  = ct * 64 + (tid & 63);
    const int sg = tid >> 6;                     // 0..3
    float acc[N_HEADS] = {};
    for (int s = sg; s < 1024; s += 4) {
        const float xv = x[((size_t)b * SEQ + s0 + s) * D_MODEL + c];
#pragma unroll
        for (int h = 0; h < N_HEADS; ++h) acc[h] += probs[h * 1024 + s] * xv;
    }

    float* xa = ws + WS_XA;
#pragma unroll
    for (int h = 0; h < N_HEADS; ++h)
        atomicAdd(&xa[((size_t)b * N_HEADS + h) * D_MODEL + c], acc[h]);
}

// ---------------------------------------------------------------------------
// K5: V[b,j] = bv[j] + sum_c xa[b, j/64, c]*wv[c,j];
//     outrow[b,j] = bo[j] + sum_c V[b,c]*wo[c,j]
// ---------------------------------------------------------------------------
__global__ __launch_bounds__(512)
void k_vo(const float* __restrict__ wv, const float* __restrict__ bv,
          const float* __restrict__ wo, const float* __restrict__ bo,
          float* __restrict__ ws) {
    __shared__ float V[D_MODEL];
    const int b = blockIdx.x;                    // 8
    const int j = threadIdx.x;                   // 512
    const float* xa = ws + WS_XA + (size_t)b * N_HEADS * D_MODEL;
    const int h = j >> 6;

    float acc = bv[j];
    for (int cc = 0; cc < D_MODEL; ++cc)
        acc += xa[h * D_MODEL + cc] * wv[cc * D_MODEL + j];
    V[j] = acc;
    __syncthreads();

    float o = bo[j];
    for (int cc = 0; cc < D_MODEL; ++cc)
        o += V[cc] * wo[cc * D_MODEL + j];
    ws[WS_OUTROW + (size_t)b * D_MODEL + j] = o;
}

// ---------------------------------------------------------------------------
// K6: out[b,l,:] = outrow[b,:]   (128 MiB coalesced float4 stream)
// ---------------------------------------------------------------------------
__global__ __launch_bounds__(256)
void k_broadcast(const float* __restrict__ ws, float* __restrict__ out) {
    const float4* rows = (const float4*)(ws + WS_OUTROW);   // [8][128]
    float4* o4 = (float4*)out;
    const size_t total  = (size_t)BATCH * SEQ * (D_MODEL / 4);
    const size_t stride = (size_t)gridDim.x * blockDim.x;
    for (size_t idx = (size_t)blockIdx.x * blockDim.x + threadIdx.x;
         idx < total; idx += stride) {
        const size_t b  = idx / ((size_t)SEQ * (D_MODEL / 4));
        const int    c4 = (int)(idx & (D_MODEL / 4 - 1));
        o4[idx] = rows[b * (D_MODEL / 4) + c4];
    }
}

// ---------------------------------------------------------------------------
extern "C" void kernel_launch(void* const* d_in, const int* in_sizes, int n_in,
                              void* d_out, int out_size, void* d_ws, size_t ws_size,
                              hipStream_t stream) {
    const float* x   = (const float*)d_in[0];
    const float* poi = (const float*)d_in[1];
    const float* wq1 = (const float*)d_in[2];
    const float* bq1 = (const float*)d_in[3];
    const float* wq2 = (const float*)d_in[4];
    const float* bq2 = (const float*)d_in[5];
    const float* wk  = (const float*)d_in[6];
    const float* bk  = (const float*)d_in[7];
    const float* wv  = (const float*)d_in[8];
    const float* bv  = (const float*)d_in[9];
    const float* wo  = (const float*)d_in[10];
    const float* bo  = (const float*)d_in[11];
    float* out = (float*)d_out;
    float* ws  = (float*)d_ws;

    k_prepare<<<1, 256, 0, stream>>>(poi, wq1, bq1, wq2, bq2, wk, bk, ws);
    k_scores<<<512, 256, 0, stream>>>(x, ws);          // 4096 wave-tiles, WMMA
    k_softmax<<<64, 256, 0, stream>>>(ws);
    dim3 gxa(8, 8, 8);
    k_xa<<<gxa, 256, 0, stream>>>(x, ws);
    k_vo<<<8, 512, 0, stream>>>(wv, bv, wo, bo, ws);
    k_broadcast<<<4096, 256, 0, stream>>>(ws, out);
}